// ChunkLayer_52664888984032
// MI455X (gfx1250) — compile-verified
//
#include <hip/hip_runtime.h>
#include <stdint.h>

#define B_ 8
#define S_ 8192
#define D_ 1024

// -------------------------------------------------------------------------
// Kernel 1: per-batch stable-partition scan over boundaries.
// One 1024-thread block per batch row; each thread owns 8 boolean bytes.
// Produces src_idx[b][slot] = source row index, and num_tokens (as float)
// into the tail of d_out.
// -------------------------------------------------------------------------
__global__ __launch_bounds__(1024) void chunk_scan_kernel(
    const uint8_t* __restrict__ boundaries,
    int* __restrict__ src_idx,
    float* __restrict__ num_tokens_out,
    int max_chunks) {
  __shared__ int lds[1024];
  const int b   = blockIdx.x;
  const int tid = threadIdx.x;

  // Load 8 boundary bytes (8B-aligned: base + b*8192 + tid*8).
  const uint2 pair = *(const uint2*)(boundaries + (size_t)b * S_ + (size_t)tid * 8);
  uint8_t v[8];
#pragma unroll
  for (int k = 0; k < 4; ++k) {
    v[k]     = (uint8_t)((pair.x >> (8 * k)) & 0xFF);
    v[4 + k] = (uint8_t)((pair.y >> (8 * k)) & 0xFF);
  }
  int cnt = 0;
#pragma unroll
  for (int k = 0; k < 8; ++k) cnt += (v[k] != 0);

  lds[tid] = cnt;
  __syncthreads();
  // Hillis-Steele inclusive scan over 1024 per-thread counts.
  for (int off = 1; off < 1024; off <<= 1) {
    int t = (tid >= off) ? lds[tid - off] : 0;
    __syncthreads();
    lds[tid] += t;
    __syncthreads();
  }
  const int incl  = lds[tid];
  const int excl  = incl - cnt;
  const int total = lds[1023];

  // Assign slots: trues -> [0, total) in order; falses -> [total, S) in order.
  int run = excl;  // number of trues strictly before current element
#pragma unroll
  for (int k = 0; k < 8; ++k) {
    const int s = tid * 8 + k;
    int slot;
    if (v[k]) {
      slot = run;
      ++run;
    } else {
      slot = total + (s - run);  // s - run == rankFalse(s)
    }
    if (slot < max_chunks) src_idx[(size_t)b * max_chunks + slot] = s;
  }

  if (tid == 0) num_tokens_out[b] = (float)total;
}

// -------------------------------------------------------------------------
// Kernel 2: bandwidth-bound row gather. One block per output row; 256 lanes
// each move 16 bytes -> exactly one 4 KB row (D=1024 f32) per block.
// Uses the CDNA5 async Global<->LDS DMA path (ASYNCcnt):
//   global_load_async_to_lds_b128  ->  s_wait_asynccnt 0
//   global_store_async_from_lds_b128
// which streams memory->LDS->memory without staging data in VGPRs.
// The builtins take v4i (vector_size(16) int) pointers per the compiler's
// own diagnostic; global side is AS(1), LDS side is AS(3).
// -------------------------------------------------------------------------
typedef int v4i __attribute__((vector_size(16)));
typedef __attribute__((address_space(1))) v4i gv4i;
typedef __attribute__((address_space(3))) v4i lv4i;

#if __has_builtin(__builtin_amdgcn_global_load_async_to_lds_b128) && \
    __has_builtin(__builtin_amdgcn_global_store_async_from_lds_b128)
#define USE_ASYNC_LDS 1
#else
#define USE_ASYNC_LDS 0
#endif

__global__ __launch_bounds__(256) void chunk_gather_kernel(
    const float* __restrict__ x,
    const int* __restrict__ src_idx,
    float* __restrict__ out,
    int max_chunks) {
  const int row = blockIdx.x;           // 0 .. B*max_chunks-1
  const int b   = row / max_chunks;
  const int tid = threadIdx.x;

  const int s = src_idx[row];
  const float* src = x + (((size_t)b * S_) + (size_t)s) * D_ + (size_t)tid * 4;
  float*       dst = out + (size_t)row * D_ + (size_t)tid * 4;

#if USE_ASYNC_LDS
  __shared__ alignas(16) char smem[256 * 16];
  char* lgen = smem + tid * 16;  // lane-private 16B staging slot
  __builtin_amdgcn_global_load_async_to_lds_b128(
      (gv4i*)(void*)src, (lv4i*)lgen, 0, 0);
  // Ensure the async load has landed in LDS before the async store reads it.
#if __has_builtin(__builtin_amdgcn_s_wait_asynccnt)
  __builtin_amdgcn_s_wait_asynccnt(0);
#else
  asm volatile("s_wait_asynccnt 0" ::: "memory");
#endif
  __builtin_amdgcn_global_store_async_from_lds_b128(
      (gv4i*)(void*)dst, (lv4i*)lgen, 0, 0);
  // implicit wait at S_ENDPGM drains the outstanding async store
#else
  const float4 t = *(const float4*)src;
  *(float4*)dst = t;
#endif
}

// -------------------------------------------------------------------------
// Launch: max_chunks is recovered from out_size = B*max_chunks*D + B.
// d_ws holds src_idx (B * max_chunks ints, <= 256 KB).
// num_tokens (second tuple output) is written as float at the tail of d_out.
// -------------------------------------------------------------------------
extern "C" void kernel_launch(void* const* d_in, const int* in_sizes, int n_in,
                              void* d_out, int out_size, void* d_ws, size_t ws_size,
                              hipStream_t stream) {
  const float*   x          = (const float*)d_in[0];
  const uint8_t* boundaries = (const uint8_t*)d_in[1];  // jax bool_: 1 byte/elem
  float*         out        = (float*)d_out;

  const int max_chunks = (out_size - B_) / (B_ * D_);
  int*   src_idx  = (int*)d_ws;
  float* num_tok  = out + (size_t)B_ * (size_t)max_chunks * (size_t)D_;

  chunk_scan_kernel<<<B_, 1024, 0, stream>>>(boundaries, src_idx, num_tok, max_chunks);
  chunk_gather_kernel<<<B_ * max_chunks, 256, 0, stream>>>(x, src_idx, out, max_chunks);
}